// DCN_4217657885255
// MI455X (gfx1250) — compile-verified
//
#include <hip/hip_runtime.h>
#include <hip/hip_bf16.h>
#include <math.h>

// ---------------------------------------------------------------------------
// DCNv2 forward for MI455X (gfx1250), fp32 via V_WMMA_F32_16X16X4_F32.
// Shapes (fixed by the reference): B=4, Cin=64, Coff=32, Cout=64, H=W=128,
// K=3, stride=1, pad=1, dil=1, DG=1.
// ---------------------------------------------------------------------------

typedef float v2f __attribute__((ext_vector_type(2)));
typedef float v8f __attribute__((ext_vector_type(8)));

#define B_    4
#define CIN   64
#define COUT  64
#define COFF  32
#define H_    128
#define W_    128
#define HW_   (H_ * W_)          // 16384
#define KK_   9
#define KOFF  (COFF * KK_)       // 288  (offset-conv GEMM K)
#define KMAIN (CIN * KK_)        // 576  (main GEMM K)
#define NCH_OM 27                // 2*9 offsets + 9 mask logits

// ---------------------------------------------------------------------------
// Kernel 0: pad the 27x288 offset-conv weight to a 32x288 A-matrix (rows
// 27..31 zero) so WMMA M-tiles are clean.
// ---------------------------------------------------------------------------
__global__ void dcn_prep_wpad(const float* __restrict__ w_off,
                              float* __restrict__ wpad) {
    int i = blockIdx.x * 256 + threadIdx.x;            // 32*288 = 9216
    if (i >= 32 * KOFF) return;
    int co = i / KOFF, k = i - co * KOFF;
    wpad[i] = (co < NCH_OM) ? w_off[co * KOFF + k] : 0.0f;
}

// ---------------------------------------------------------------------------
// Kernel 1: offset/mask conv as implicit GEMM.
//   M=32 (27 live), K=288, N = B*HW = 65536.  N-tile 32, 4 waves per block.
// ---------------------------------------------------------------------------
__global__ __launch_bounds__(128)
void dcn_offset_conv(const float* __restrict__ xoff,   // [B,32,H,W]
                     const float* __restrict__ wpad,   // [32,288]
                     const float* __restrict__ boff,   // [27]
                     float* __restrict__ om) {         // [B,27,H,W]
    __shared__ float cols[72 * 33];                    // 8 ch * 9 taps, pad 33

    const int tid  = threadIdx.x;
    const int lane = tid & 31;          // wave32
    const int wid  = tid >> 5;          // 0..3
    const int half = lane >> 4;         // 0/1
    const int lm   = lane & 15;
    const int m0   = (wid >> 1) * 16;   // 0 or 16
    const int ns   = (wid & 1) * 16;    // 0 or 16

    const int n0  = blockIdx.x * 32;    // flat over B*HW (32 | HW)
    const int b   = n0 / HW_;
    const int sp0 = n0 - b * HW_;

    v8f acc;
#pragma unroll
    for (int r = 0; r < 8; ++r) {
        int M = m0 + r + 8 * half;
        acc[r] = (M < NCH_OM) ? boff[M] : 0.0f;
    }

    for (int c0 = 0; c0 < COFF; c0 += 8) {
        __syncthreads();
        // stage im2col slice: 8*9*32 = 2304 entries
        for (int e = tid; e < 2304; e += 128) {
            int cl  = e / 288;
            int rem = e - cl * 288;
            int kk  = rem >> 5;
            int nl  = rem & 31;
            int sp  = sp0 + nl;
            int ho  = sp >> 7, wo = sp & (W_ - 1);
            int y   = ho + (kk / 3) - 1;
            int x   = wo + (kk % 3) - 1;
            float v = 0.0f;
            if (y >= 0 && y < H_ && x >= 0 && x < W_)
                v = xoff[(b * COFF + c0 + cl) * HW_ + y * W_ + x];
            cols[(cl * KK_ + kk) * 33 + nl] = v;
        }
        __syncthreads();
#pragma unroll 2
        for (int kl = 0; kl < 72; kl += 4) {
            int kg = c0 * KK_ + kl;
            v2f a, bb;
            const float* ap = wpad + (m0 + lm) * KOFF + kg + 2 * half;
            a.x = ap[0];  a.y = ap[1];
            const float* bp = cols + (kl + 2 * half) * 33 + ns + lm;
            bb.x = bp[0]; bb.y = bp[33];
            acc = __builtin_amdgcn_wmma_f32_16x16x4_f32(
                false, a, false, bb, (short)0, acc, false, false);
        }
    }

    const int spn = sp0 + ns + lm;
#pragma unroll
    for (int r = 0; r < 8; ++r) {
        int M = m0 + r + 8 * half;
        if (M < NCH_OM) om[(b * NCH_OM + M) * HW_ + spn] = acc[r];
    }
}

// ---------------------------------------------------------------------------
// Kernel 2: fused bilinear gather + main GEMM.
//   M=64, K=576, N=65536.  N-tile 32, 8 waves per block (4 M-tiles x 2 N-sub).
// ---------------------------------------------------------------------------
__global__ __launch_bounds__(256)
void dcn_main(const float* __restrict__ x,      // [B,64,H,W]
              const float* __restrict__ om,     // [B,27,H,W]
              const float* __restrict__ wmat,   // [64,576] = weight flat
              const float* __restrict__ bias,   // [64]
              float* __restrict__ out) {        // [B,64,H,W]
    __shared__ float pw[4][288];                // bilinear weights (mask-folded)
    __shared__ int   pi[4][288];                // clamped gather indices
    __shared__ float cols[72 * 33];

    const int tid  = threadIdx.x;
    const int lane = tid & 31;
    const int wid  = tid >> 5;          // 0..7
    const int half = lane >> 4;
    const int lm   = lane & 15;
    const int m0   = (wid >> 1) * 16;   // 0,16,32,48
    const int ns   = (wid & 1) * 16;

    const int n0  = blockIdx.x * 32;
    const int b   = n0 / HW_;
    const int sp0 = n0 - b * HW_;

    // ---- per-tile sampling descriptors: 9 taps * 32 pixels ----
    for (int e = tid; e < 288; e += 256) {
        int kk = e >> 5, nl = e & 31;
        int sp = sp0 + nl;
        int ho = sp >> 7, wo = sp & (W_ - 1);
        const float* omb = om + b * NCH_OM * HW_ + sp;
        float oy = omb[(2 * kk) * HW_];
        float ox = omb[(2 * kk + 1) * HW_];
        float mz = omb[(18 + kk) * HW_];
        float m  = 1.0f / (1.0f + __expf(-mz));          // sigmoid modulation
        float py = (float)(ho - 1 + kk / 3) + oy;
        float px = (float)(wo - 1 + kk % 3) + ox;
        float y0f = floorf(py), x0f = floorf(px);
        float wy = py - y0f, wx = px - x0f;
        int y0 = (int)y0f, x0 = (int)x0f;
        int y1 = y0 + 1,  x1 = x0 + 1;
        bool vy0 = (y0 >= 0) && (y0 < H_), vy1 = (y1 >= 0) && (y1 < H_);
        bool vx0 = (x0 >= 0) && (x0 < W_), vx1 = (x1 >= 0) && (x1 < W_);
        int cy0 = min(max(y0, 0), H_ - 1), cy1 = min(max(y1, 0), H_ - 1);
        int cx0 = min(max(x0, 0), W_ - 1), cx1 = min(max(x1, 0), W_ - 1);
        pw[0][e] = (vy0 && vx0) ? (1.0f - wy) * (1.0f - wx) * m : 0.0f;
        pw[1][e] = (vy0 && vx1) ? (1.0f - wy) * wx * m          : 0.0f;
        pw[2][e] = (vy1 && vx0) ? wy * (1.0f - wx) * m          : 0.0f;
        pw[3][e] = (vy1 && vx1) ? wy * wx * m                   : 0.0f;
        pi[0][e] = cy0 * W_ + cx0;
        pi[1][e] = cy0 * W_ + cx1;
        pi[2][e] = cy1 * W_ + cx0;
        pi[3][e] = cy1 * W_ + cx1;
    }

    v8f acc;
#pragma unroll
    for (int r = 0; r < 8; ++r)
        acc[r] = bias[m0 + r + 8 * half];

    for (int c0 = 0; c0 < CIN; c0 += 8) {
        __syncthreads();   // descriptors ready / previous chunk consumed
        // gather-sample 8 channels worth of cols: 2304 entries
        for (int e = tid; e < 2304; e += 256) {
            int cl  = e / 288;
            int rem = e - cl * 288;
            const float* xp = x + (b * CIN + c0 + cl) * HW_;
            float v = pw[0][rem] * xp[pi[0][rem]]
                    + pw[1][rem] * xp[pi[1][rem]]
                    + pw[2][rem] * xp[pi[2][rem]]
                    + pw[3][rem] * xp[pi[3][rem]];
            int kk = rem >> 5, nl = rem & 31;
            cols[(cl * KK_ + kk) * 33 + nl] = v;
        }
        __syncthreads();
#pragma unroll 2
        for (int kl = 0; kl < 72; kl += 4) {
            int kg = c0 * KK_ + kl;
            v2f a, bb;
            const float* ap = wmat + (m0 + lm) * KMAIN + kg + 2 * half;
            a.x = ap[0];  a.y = ap[1];
            const float* bp = cols + (kl + 2 * half) * 33 + ns + lm;
            bb.x = bp[0]; bb.y = bp[33];
            acc = __builtin_amdgcn_wmma_f32_16x16x4_f32(
                false, a, false, bb, (short)0, acc, false, false);
        }
    }

    const int spn = sp0 + ns + lm;
#pragma unroll
    for (int r = 0; r < 8; ++r) {
        int M = m0 + r + 8 * half;
        out[(b * COUT + M) * HW_ + spn] = acc[r];
    }
}

// ---------------------------------------------------------------------------
// Launch: prep -> offset conv -> fused deformable conv (stream-ordered).
// Workspace: om [4*27*16384] floats, then wpad [32*288] floats (~7.1 MB).
// ---------------------------------------------------------------------------
extern "C" void kernel_launch(void* const* d_in, const int* in_sizes, int n_in,
                              void* d_out, int out_size, void* d_ws, size_t ws_size,
                              hipStream_t stream) {
    (void)in_sizes; (void)n_in; (void)out_size; (void)ws_size;
    const float* x    = (const float*)d_in[0];   // [4,64,128,128]
    const float* xoff = (const float*)d_in[1];   // [4,32,128,128]
    const float* woff = (const float*)d_in[2];   // [27,32,3,3]
    const float* boff = (const float*)d_in[3];   // [27]
    const float* w    = (const float*)d_in[4];   // [64,64,3,3] -> [64,576]
    const float* bias = (const float*)d_in[5];   // [64]
    float* out  = (float*)d_out;

    float* om   = (float*)d_ws;                  // 4*27*16384 floats
    float* wpad = om + (size_t)B_ * NCH_OM * HW_;

    const int n_total = B_ * HW_;                // 65536
    dcn_prep_wpad<<<(32 * KOFF + 255) / 256, 256, 0, stream>>>(woff, wpad);
    dcn_offset_conv<<<n_total / 32, 128, 0, stream>>>(xoff, wpad, boff, om);
    dcn_main<<<n_total / 32, 256, 0, stream>>>(x, om, w, bias, out);
}